// YOLOv3Loss_16003048145236
// MI455X (gfx1250) — compile-verified
//
#include <hip/hip_runtime.h>
#include <math.h>

// ---------------- constants from the reference ----------------
#define K_NC   80
#define K_NA   3
#define K_B    32
#define K_NT   512
#define K_M    (5 * K_NA * K_NT)   // 7680 rows per level
#define K_CP   0.95f
#define K_CN   0.05f
#define K_EPS  1e-7f
#define K_4PI2 0.4052847345693511f // 4/pi^2

typedef float v2f __attribute__((ext_vector_type(2)));
typedef float v8f __attribute__((ext_vector_type(8)));

// ---- exact 32-lane sum via V_WMMA_F32_16X16X4_F32 with all-ones B ----
// A layout (16x4 f32): lanes 0-15 -> M=lane, K0 in a[0], K1 in a[1];
// lanes 16-31 -> M=lane-16, K2 in a[0], K3 in a[1].
// With a = {p, 0} and B == ones:  D[m,n] = p_m + p_{m+16}  (any n).
// C/D layout: VGPR r holds M=r (lanes 0-15) / M=r+8 (lanes 16-31).
// => sum_r d[r] at lane0 covers M=0..7, at lane16 covers M=8..15.
// Combine the two halves with v_readlane (constant lanes -> SGPR, no LDS).
__device__ __forceinline__ float wave_sum32(float p) {
    v2f a; a[0] = p;    a[1] = 0.0f;
    v2f b; b[0] = 1.0f; b[1] = 1.0f;     // all-ones B: layout independent
    v8f c = {};
    c = __builtin_amdgcn_wmma_f32_16x16x4_f32(false, a, false, b,
                                              (short)0, c, false, false);
    float s = c[0] + c[1] + c[2] + c[3] + c[4] + c[5] + c[6] + c[7];
    int lo = __builtin_amdgcn_readlane(__float_as_int(s), 0);
    int hi = __builtin_amdgcn_readlane(__float_as_int(s), 16);
    return __int_as_float(lo) + __int_as_float(hi);
}

__device__ __forceinline__ float sigmoidf_(float x) {
    return 1.0f / (1.0f + expf(-x));
}
__device__ __forceinline__ float bce_logits_(float x, float y) {
    return fmaxf(x, 0.0f) - x * y + log1pf(expf(-fabsf(x)));
}

__device__ __forceinline__ float ciou_f(float px, float py, float pw, float ph,
                                        float tx, float ty, float tw, float th) {
    float b1x1 = px - pw * 0.5f, b1x2 = px + pw * 0.5f;
    float b1y1 = py - ph * 0.5f, b1y2 = py + ph * 0.5f;
    float b2x1 = tx - tw * 0.5f, b2x2 = tx + tw * 0.5f;
    float b2y1 = ty - th * 0.5f, b2y2 = ty + th * 0.5f;
    float iw = fmaxf(fminf(b1x2, b2x2) - fmaxf(b1x1, b2x1), 0.0f);
    float ih = fmaxf(fminf(b1y2, b2y2) - fmaxf(b1y1, b2y1), 0.0f);
    float inter = iw * ih;
    float uni   = pw * ph + tw * th - inter + K_EPS;
    float iou   = inter / uni;
    float cw = fmaxf(b1x2, b2x2) - fminf(b1x1, b2x1);
    float ch = fmaxf(b1y2, b2y2) - fminf(b1y1, b2y1);
    float c2 = cw * cw + ch * ch + K_EPS;
    float dx = b2x1 + b2x2 - b1x1 - b1x2;
    float dy = b2y1 + b2y2 - b1y1 - b1y2;
    float rho2 = (dx * dx + dy * dy) * 0.25f;
    float dv = atanf(tw / th) - atanf(pw / ph);
    float v  = K_4PI2 * dv * dv;
    float alpha = v / (v - iou + (1.0f + K_EPS));
    return iou - (rho2 / c2 + v * alpha);
}

// ---------------- workspace zero ----------------
__global__ void zero_ws_kernel(float* __restrict__ p, int n) {
    int i = blockIdx.x * blockDim.x + threadIdx.x;
    if (i < n) p[i] = 0.0f;
}

// ---------------- per-level row kernel: build_level + gather losses ----------------
// acc: [lbox_sum, lcls_sum, nvalid, lobj_sum] for this level
__global__ void level_gather_kernel(const float* __restrict__ pi,
                                    const float* __restrict__ tg,
                                    float* __restrict__ tobj,
                                    float* __restrict__ acc,
                                    int W, int H,
                                    float a0w, float a0h,
                                    float a1w, float a1h,
                                    float a2w, float a2h) {
    int m = blockIdx.x * blockDim.x + threadIdx.x;
    float lbox_p = 0.0f, lcls_p = 0.0f, nval_p = 0.0f;
    if (m < K_M) {
        int o   = m / (K_NA * K_NT);
        int idx = m - o * (K_NA * K_NT);
        int a   = idx >> 9;          // / 512
        int n   = idx & 511;
        const float* t = tg + (size_t)n * 6;
        float bimg = t[0], clsf = t[1];
        float x1 = t[2], y1 = t[3], x2 = t[4], y2 = t[5];
        float gx = (x1 + x2) * 0.5f * (float)W;
        float gy = (y1 + y2) * 0.5f * (float)H;
        float gw = (x2 - x1) * (float)W;
        float gh = (y2 - y1) * (float)H;
        float aw = (a == 0) ? a0w : ((a == 1) ? a1w : a2w);
        float ah = (a == 0) ? a0h : ((a == 1) ? a1h : a2h);
        float rx = gw / aw, ry = gh / ah;
        float mr = fmaxf(fmaxf(rx, 1.0f / rx), fmaxf(ry, 1.0f / ry));
        bool amask = mr < 4.0f;
        float gix = (float)W - gx, giy = (float)H - gy;
        bool jkx = ((gx  - floorf(gx))  < 0.5f) && (gx  > 1.0f);
        bool jky = ((gy  - floorf(gy))  < 0.5f) && (gy  > 1.0f);
        bool lmx = ((gix - floorf(gix)) < 0.5f) && (gix > 1.0f);
        bool lmy = ((giy - floorf(giy)) < 0.5f) && (giy > 1.0f);
        bool om  = (o == 0) ? true
                 : (o == 1) ? jkx
                 : (o == 2) ? jky
                 : (o == 3) ? lmx : lmy;
        bool valid = om && amask;
        const float offx_t[5] = {0.0f, 0.5f, 0.0f, -0.5f, 0.0f};
        const float offy_t[5] = {0.0f, 0.0f, 0.5f, 0.0f, -0.5f};
        float gijx = floorf(gx - offx_t[o]);
        float gijy = floorf(gy - offy_t[o]);
        int gi = min(max((int)gijx, 0), W - 1);
        int gj = min(max((int)gijy, 0), H - 1);
        int b  = (int)bimg;
        int c  = (int)clsf;
        if (valid) {
            size_t cell = (((size_t)b * K_NA + a) * H + gj) * W + gi;
            const float* ps = pi + cell * 85;
            float pxc = sigmoidf_(ps[0]) * 2.0f - 0.5f;
            float pyc = sigmoidf_(ps[1]) * 2.0f - 0.5f;
            float sw  = sigmoidf_(ps[2]) * 2.0f;
            float sh  = sigmoidf_(ps[3]) * 2.0f;
            float pw  = sw * sw * aw;
            float ph  = sh * sh * ah;
            float tbx = gx - gijx, tby = gy - gijy;
            float iou = ciou_f(pxc, pyc, pw, ph, tbx, tby, gw, gh);
            lbox_p = 1.0f - iou;
            nval_p = 1.0f;
            float score = fmaxf(iou, 0.0f);
            if (score > 0.0f) {
                // non-negative floats: uint bit order == float order
                atomicMax((unsigned int*)(tobj + cell), __float_as_uint(score));
            }
            float s = 0.0f;
            #pragma unroll 4
            for (int k = 0; k < K_NC; ++k) {
                float x = ps[5 + k];
                float y = (k == c) ? K_CP : K_CN;
                s += bce_logits_(x, y);
            }
            lcls_p = s;
        }
    }
    // reconverged: full-EXEC wave reductions via v_wmma_f32_16x16x4_f32
    float lbox_s = wave_sum32(lbox_p);
    float lcls_s = wave_sum32(lcls_p);
    float nval_s = wave_sum32(nval_p);
    if ((threadIdx.x & 31) == 0) {
        atomicAdd(&acc[0], lbox_s);
        atomicAdd(&acc[1], lcls_s);
        atomicAdd(&acc[2], nval_s);
    }
}

// ---------------- objectness BCE over every cell (bandwidth-bound) ----------------
__global__ void obj_loss_kernel(const float* __restrict__ pi,
                                const float* __restrict__ tobj,
                                float* __restrict__ acc, int ncells) {
    int i = blockIdx.x * blockDim.x + threadIdx.x;
    float p = 0.0f;
    if (i < ncells) {
        float x = pi[(size_t)i * 85 + 4];
        float t = tobj[i];
        p = bce_logits_(x, t);
    }
    float s = wave_sum32(p);
    if ((threadIdx.x & 31) == 0) atomicAdd(&acc[3], s);
}

// ---------------- finalize ----------------
__global__ void finalize_kernel(const float* __restrict__ acc,
                                float* __restrict__ out) {
    const float balance[3] = {4.0f, 1.0f, 0.4f};
    const float ncells[3]  = {(float)(K_B * K_NA * 52 * 52),
                              (float)(K_B * K_NA * 26 * 26),
                              (float)(K_B * K_NA * 13 * 13)};
    float lbox = 0.0f, lobj = 0.0f, lcls = 0.0f;
    for (int i = 0; i < 3; ++i) {
        float nv = fmaxf(acc[i * 4 + 2], 1.0f);
        lbox += acc[i * 4 + 0] / nv;
        lcls += acc[i * 4 + 1] / (nv * (float)K_NC);
        lobj += balance[i] * acc[i * 4 + 3] / ncells[i];
    }
    lbox *= 0.05f;   // BOX_W
    lcls *= 0.5f;    // CLS_W
    // OBJ_W = 1
    float total = (lbox + lobj + lcls) * (float)K_B;
    out[0] = total; out[1] = lbox; out[2] = lobj; out[3] = lcls;
}

extern "C" void kernel_launch(void* const* d_in, const int* in_sizes, int n_in,
                              void* d_out, int out_size, void* d_ws, size_t ws_size,
                              hipStream_t stream) {
    const float* p[3] = {(const float*)d_in[0], (const float*)d_in[1],
                         (const float*)d_in[2]};
    const float* tg = (const float*)d_in[3];
    float* out = (float*)d_out;
    float* ws  = (float*)d_ws;

    const int Wh[3]    = {52, 26, 13};
    const int cells[3] = {K_B * K_NA * 52 * 52,
                          K_B * K_NA * 26 * 26,
                          K_B * K_NA * 13 * 13};
    // anchors = PIX / stride
    const float anc[3][6] = {
        {1.25f,    1.625f,  2.0f,    3.75f,   4.125f,    2.875f },
        {1.875f,   3.8125f, 3.875f,  2.8125f, 3.6875f,   7.4375f},
        {3.625f,   2.8125f, 4.875f,  6.1875f, 11.65625f, 10.1875f}
    };

    float* acc = ws;                       // 3 levels x 4 accumulators
    float* tobj[3];
    tobj[0] = ws + 16;
    tobj[1] = tobj[0] + cells[0];
    tobj[2] = tobj[1] + cells[1];
    int ws_floats = 16 + cells[0] + cells[1] + cells[2];

    zero_ws_kernel<<<(ws_floats + 255) / 256, 256, 0, stream>>>(ws, ws_floats);

    for (int i = 0; i < 3; ++i) {
        level_gather_kernel<<<(K_M + 255) / 256, 256, 0, stream>>>(
            p[i], tg, tobj[i], acc + i * 4, Wh[i], Wh[i],
            anc[i][0], anc[i][1], anc[i][2], anc[i][3], anc[i][4], anc[i][5]);
    }
    for (int i = 0; i < 3; ++i) {
        obj_loss_kernel<<<(cells[i] + 255) / 256, 256, 0, stream>>>(
            p[i], tobj[i], acc + i * 4, cells[i]);
    }
    finalize_kernel<<<1, 1, 0, stream>>>(acc, out);
    (void)in_sizes; (void)n_in; (void)out_size; (void)ws_size;
}